// MFA_69758858821990
// MI455X (gfx1250) — compile-verified
//
#include <hip/hip_runtime.h>
#include <math.h>

// ---------------------------------------------------------------------------
// MFA per-component log-likelihood for MI455X (gfx1250), fp32 WMMA path.
//   K=128 components, D=1024 features, L=16 factors, N=16384 samples.
// out[n,k] = PI[k] - 0.5*(D*log2pi + logdet(Lm_k) - sum(log iD_k) + c1_k)
//            - 0.5*(x^2 . iD_k) + (x . iD_k*MU_k) + 0.5*quad(n,k)
// quad = y^T iL y,  y = x.(iD*A)_k - muproj_k
// ---------------------------------------------------------------------------

typedef float v2f __attribute__((ext_vector_type(2)));
typedef float v4f __attribute__((ext_vector_type(4)));
typedef float v8f __attribute__((ext_vector_type(8)));

#define KCOMP 128
#define DFEAT 1024
#define LFAC  16
#define NROWS 16384

// workspace layout (float offsets)
#define OFF_WY   0u          // [k][i2<128][lane<32][4]  : iD*A, B-operand swizzled (2,097,152)
#define OFF_WQ   2097152u    // [kg<8][i2<128][lane][4]  : -0.5*iD                 (131,072)
#define OFF_WM   2228224u    // [kg<8][i2<128][lane][4]  : +iD*MU                  (131,072)
#define OFF_IL   2359296u    // [k][s2<2][lane][4]       : iL B-operand            (32,768)
#define OFF_MP   2392064u    // [k][l<16]                : muproj                  (2,048)
#define OFF_CST  2394112u    // [k]                      : folded constant         (128)

#define LOG2PI 1.8378770664093453f

static __device__ __forceinline__ v8f wmma_f32(v2f a, v2f b, v8f c) {
  return __builtin_amdgcn_wmma_f32_16x16x4_f32(false, a, false, b, (short)0, c,
                                               false, false);
}

// ---------------------------------------------------------------------------
// Prep kernel: one block per component k.
// ---------------------------------------------------------------------------
__global__ __launch_bounds__(256) void mfa_prep(const float* __restrict__ PI,
                                                const float* __restrict__ MU,
                                                const float* __restrict__ A,
                                                const float* __restrict__ Dm,
                                                float* __restrict__ ws) {
  __shared__ float iDs[DFEAT];
  __shared__ float Lm[16 * 16];
  __shared__ float iLm[16 * 16];
  __shared__ float red[256];
  const int k = blockIdx.x;
  const int tid = threadIdx.x;
  const int kg = k >> 4, kc = k & 15;

  float slog = 0.0f, c1 = 0.0f;
  for (int d = tid; d < DFEAT; d += 256) {
    float Dv = Dm[k * DFEAT + d];
    float id = 1.0f / (Dv * Dv);
    iDs[d] = id;
    float mu = MU[k * DFEAT + d];
    slog += logf(id);
    c1 += id * mu * mu;

    int d4 = d >> 2, sub = d & 3;
    int i2 = d4 >> 1;
    int j = sub & 1;
    int chalf = 16 * (sub >> 1);
    int q = (d4 & 1) * 2 + j;
    // term1 B tiles: column = component-within-group
    int idx = ((kg * 128 + i2) * 32 + (kc + chalf)) * 4 + q;
    ws[OFF_WQ + idx] = -0.5f * id;
    ws[OFF_WM + idx] = id * mu;
    // y B tiles: column = factor l
    for (int l = 0; l < LFAC; ++l) {
      int idy = ((k * 128 + i2) * 32 + (l + chalf)) * 4 + q;
      ws[OFF_WY + idy] = id * A[(k * DFEAT + d) * LFAC + l];
    }
  }

  // block reductions for sum(log iD) and c1
  red[tid] = slog;
  __syncthreads();
  for (int s = 128; s > 0; s >>= 1) {
    if (tid < s) red[tid] += red[tid + s];
    __syncthreads();
  }
  float slogT = red[0];
  __syncthreads();
  red[tid] = c1;
  __syncthreads();
  for (int s = 128; s > 0; s >>= 1) {
    if (tid < s) red[tid] += red[tid + s];
    __syncthreads();
  }
  float c1T = red[0];
  __syncthreads();

  // Lm = I + A^T diag(iD) A  -> one (l,m) entry per thread
  {
    int l = tid >> 4, m = tid & 15;
    float s = (l == m) ? 1.0f : 0.0f;
    for (int d = 0; d < DFEAT; ++d)
      s += A[(k * DFEAT + d) * LFAC + l] * iDs[d] *
           A[(k * DFEAT + d) * LFAC + m];
    Lm[tid] = s;
  }
  // muproj[l] = sum_d MU*iD*A[:,l]
  __syncthreads();
  if (tid < 16) {
    float s = 0.0f;
    for (int d = 0; d < DFEAT; ++d)
      s += MU[k * DFEAT + d] * iDs[d] * A[(k * DFEAT + d) * LFAC + tid];
    ws[OFF_MP + k * 16 + tid] = s;
  }
  __syncthreads();

  // Gauss-Jordan inverse + logdet on SPD 16x16 (single lane; K=128 blocks)
  if (tid == 0) {
    for (int i = 0; i < 256; ++i) iLm[i] = ((i >> 4) == (i & 15)) ? 1.0f : 0.0f;
    float logdet = 0.0f;
    for (int p = 0; p < 16; ++p) {
      float piv = Lm[p * 16 + p];
      logdet += logf(piv);
      float inv = 1.0f / piv;
      for (int j = 0; j < 16; ++j) {
        Lm[p * 16 + j] *= inv;
        iLm[p * 16 + j] *= inv;
      }
      for (int r = 0; r < 16; ++r) {
        if (r == p) continue;
        float f = Lm[r * 16 + p];
        for (int j = 0; j < 16; ++j) {
          Lm[r * 16 + j] -= f * Lm[p * 16 + j];
          iLm[r * 16 + j] -= f * iLm[p * 16 + j];
        }
      }
    }
    ws[OFF_CST + k] =
        PI[k] - 0.5f * (1024.0f * LOG2PI + logdet - slogT + c1T);
  }
  __syncthreads();

  // iL into B-operand swizzle: [k][s2][lane][4]
  {
    int t = tid;             // 0..255 = 2*32*4
    int s2 = t >> 7;
    int rem = t & 127;
    int c = rem >> 2;
    int q = rem & 3;
    int s = 2 * s2 + (q >> 1);
    int j = q & 1;
    int rowi = 4 * s + 2 * (c >> 4) + j;
    int col = c & 15;
    ws[OFF_IL + k * 256 + t] = iLm[rowi * 16 + col];
  }
}

// ---------------------------------------------------------------------------
// Main kernel: 256 blocks x 256 threads (8 wave32).
// Block: 64 rows of x staged in LDS via async-to-LDS; wave w -> row-tile
// (w&3), k-half (w>>2).
// ---------------------------------------------------------------------------
__global__ __launch_bounds__(256) void mfa_main(const float* __restrict__ x,
                                                const float* __restrict__ ws,
                                                float* __restrict__ out) {
  extern __shared__ float smem[];
  float* xs = smem;               // [256][64][4] interleaved x tile (256 KB)
  float* ys = smem + 65536;       // per-wave 16x18 scratch (Y, then Y*Z)

  const int tid = threadIdx.x;
  const int lane = tid & 31;
  const int wave = tid >> 5;
  const int rowBase = blockIdx.x * 64;

  // ---- stage x tile with GLOBAL_LOAD_ASYNC_TO_LDS_B128 (ASYNCcnt path) ----
  for (int it = 0; it < 64; ++it) {
    int flat = it * 256 + tid;
    int d4 = flat & 255;
    int row = flat >> 8;
    const float* gp = x + (size_t)(rowBase + row) * DFEAT + d4 * 4;
    unsigned lds_off = (unsigned)(size_t)(xs + (d4 * 64 + row) * 4);
    asm volatile("global_load_async_to_lds_b128 %0, %1, off"
                 :
                 : "v"(lds_off), "v"(gp)
                 : "memory");
  }
  asm volatile("s_wait_asynccnt 0x0" ::: "memory");
  __syncthreads();

  const int rt = wave & 3;
  const int khalf = wave >> 2;
  const int ln16 = lane & 15;
  const int hi = lane >> 4;                 // 0/1 : which d-pair of the step
  float* yw = ys + wave * (16 * 18);
  const float* xl = xs + (rt * 16 + ln16) * 4 + hi * 2;

  for (int kg = khalf * 4; kg < khalf * 4 + 4; ++kg) {
    v8f acc = {};                           // output tile accumulator (n x kcol)

    for (int kc = 0; kc < 16; ++kc) {
      const int k = kg * 16 + kc;
      const float* wyk = ws + OFF_WY + (size_t)k * 16384 + lane * 4;
      __builtin_prefetch(wyk + 16384, 0, 1);   // next component's B stream

      // ---- y = x . (iD*A)_k : 256 WMMAs, B-stream pipelined 2 iters ahead --
      v8f yacc = {};
      auto ystep = [&](int i, v4f b) {
        v2f a0 = *(const v2f*)(xl + (2 * i) * 256);
        v2f a1 = *(const v2f*)(xl + (2 * i + 1) * 256);
        yacc = wmma_f32(a0, (v2f){b.x, b.y}, yacc);
        yacc = wmma_f32(a1, (v2f){b.z, b.w}, yacc);
      };
      v4f c0 = *(const v4f*)(wyk);
      v4f c1 = *(const v4f*)(wyk + 128);
#pragma unroll 2
      for (int i2 = 0; i2 < 126; i2 += 2) {
        v4f n0 = *(const v4f*)(wyk + (i2 + 2) * 128);
        v4f n1 = *(const v4f*)(wyk + (i2 + 3) * 128);
        ystep(i2, c0);
        ystep(i2 + 1, c1);
        c0 = n0;
        c1 = n1;
      }
      ystep(126, c0);
      ystep(127, c1);

      // subtract mu-projection (per column l = ln16)
      float mp = ws[OFF_MP + k * 16 + ln16];
#pragma unroll
      for (int r = 0; r < 8; ++r) yacc[r] -= mp;

      // Y (D-layout) -> LDS so we can refetch it in A-layout
#pragma unroll
      for (int r = 0; r < 8; ++r) yw[(r + 8 * hi) * 18 + ln16] = yacc[r];

      // ---- Z = Y * iL : 4 WMMA steps ----
      v8f zacc = {};
      const float* ilk = ws + OFF_IL + k * 256 + lane * 4;
#pragma unroll
      for (int s2 = 0; s2 < 2; ++s2) {
        v4f bb = *(const v4f*)(ilk + s2 * 128);
        v2f a0 = *(const v2f*)(yw + ln16 * 18 + (8 * s2 + 2 * hi));
        v2f a1 = *(const v2f*)(yw + ln16 * 18 + (8 * s2 + 4 + 2 * hi));
        zacc = wmma_f32(a0, (v2f){bb.x, bb.y}, zacc);
        zacc = wmma_f32(a1, (v2f){bb.z, bb.w}, zacc);
      }

      // ---- 0.5*quad into column kc via 4 more WMMAs: P=(Y.Z), B=E ----
      // P -> LDS (same scratch; DS ops from one wave are in-order)
#pragma unroll
      for (int r = 0; r < 8; ++r)
        yw[(r + 8 * hi) * 18 + ln16] = yacc[r] * zacc[r];
      float sel = (ln16 == kc) ? 0.5f : 0.0f;
      v2f e = {sel, sel};
#pragma unroll
      for (int s = 0; s < 4; ++s) {
        v2f a0 = *(const v2f*)(yw + ln16 * 18 + (4 * s + 2 * hi));
        acc = wmma_f32(a0, e, acc);
      }
    }

    // ---- term1 tiles: (-0.5*iD).x^2 + (iD*MU).x, pipelined B streams ----
    {
      const float* wq = ws + OFF_WQ + (size_t)kg * 16384 + lane * 4;
      const float* wm = ws + OFF_WM + (size_t)kg * 16384 + lane * 4;
      auto t1step = [&](int i, v4f bq, v4f bm) {
        v2f a0 = *(const v2f*)(xl + (2 * i) * 256);
        v2f a1 = *(const v2f*)(xl + (2 * i + 1) * 256);
        v2f a0s = {a0.x * a0.x, a0.y * a0.y};
        v2f a1s = {a1.x * a1.x, a1.y * a1.y};
        acc = wmma_f32(a0s, (v2f){bq.x, bq.y}, acc);
        acc = wmma_f32(a1s, (v2f){bq.z, bq.w}, acc);
        acc = wmma_f32(a0, (v2f){bm.x, bm.y}, acc);
        acc = wmma_f32(a1, (v2f){bm.z, bm.w}, acc);
      };
      v4f q0 = *(const v4f*)(wq);
      v4f m0 = *(const v4f*)(wm);
#pragma unroll 2
      for (int i2 = 0; i2 < 127; ++i2) {
        v4f qn = *(const v4f*)(wq + (i2 + 1) * 128);
        v4f mn = *(const v4f*)(wm + (i2 + 1) * 128);
        t1step(i2, q0, m0);
        q0 = qn;
        m0 = mn;
      }
      t1step(127, q0, m0);
    }

    // epilogue: add folded constant, store (n, k) tile
    float cst = ws[OFF_CST + kg * 16 + ln16];
#pragma unroll
    for (int r = 0; r < 8; ++r) {
      int n = rowBase + rt * 16 + r + 8 * hi;
      out[(size_t)n * KCOMP + kg * 16 + ln16] = acc[r] + cst;
    }
  }
}

// ---------------------------------------------------------------------------
extern "C" void kernel_launch(void* const* d_in, const int* in_sizes, int n_in,
                              void* d_out, int out_size, void* d_ws,
                              size_t ws_size, hipStream_t stream) {
  (void)in_sizes; (void)n_in; (void)out_size; (void)ws_size;
  const float* x  = (const float*)d_in[0];
  const float* PI = (const float*)d_in[1];
  const float* MU = (const float*)d_in[2];
  const float* A  = (const float*)d_in[3];
  const float* Dm = (const float*)d_in[4];
  float* out = (float*)d_out;
  float* ws  = (float*)d_ws;

  mfa_prep<<<KCOMP, 256, 0, stream>>>(PI, MU, A, Dm, ws);

  const size_t smem = (65536 + 8 * 16 * 18) * sizeof(float);  // 271,360 B
  hipFuncSetAttribute((const void*)mfa_main,
                      hipFuncAttributeMaxDynamicSharedMemorySize, (int)smem);
  mfa_main<<<NROWS / 64, 256, smem, stream>>>(x, ws, out);
}